// EdgeInit_53730040873189
// MI455X (gfx1250) — compile-verified
//
#include <hip/hip_runtime.h>
#include <hip/hip_bf16.h>

// ---------------------------------------------------------------------------
// EdgeInit for MI455X (gfx1250): relu(x@W+b) -> edge avg -> batchnorm(edges)
// Stage 1 uses V_WMMA_F32_16X16X4_F32 (exact fp32 matrix pipe).
// Stages 2-4 are bandwidth-bound; h (12.8MB) stays L2-resident for gathers.
// ---------------------------------------------------------------------------

typedef float v2f __attribute__((ext_vector_type(2)));
typedef float v8f __attribute__((ext_vector_type(8)));

#define D_IN   256
#define D_OUT  64
#define NB_RED 1024   // reduction blocks -> NB_RED*128 partial floats

// ---------------- Kernel 1: h = relu(x @ W + b) via f32 WMMA ----------------
// One wave (32 lanes) computes one 16x16 tile of h. K looped in steps of 4.
// A (16x4 f32): lanes 0-15 hold row M=lane, VGPR0=K, VGPR1=K+1;
//               lanes 16-31 hold row M=lane-16, VGPR0=K+2, VGPR1=K+3.
// B (4x16 f32): mirrored, lane = output column.
// C/D (16x16 f32, 8 VGPRs): VGPR j -> row j (lanes 0-15) / row j+8 (16-31).
__global__ __launch_bounds__(32)
void EdgeInit_gemm_relu(const float* __restrict__ x,
                        const float* __restrict__ W,
                        const float* __restrict__ bias,
                        float* __restrict__ h) {
    const int lane = threadIdx.x;
    const int half = lane >> 4;      // 0 or 1
    const int lm   = lane & 15;
    const int m0   = blockIdx.x * 16;
    const int n0   = blockIdx.y * 16;

    const float* xrow = x + (size_t)(m0 + lm) * D_IN;
    const float* wcol = W + (n0 + lm);

    v8f c = {};
    #pragma unroll 8
    for (int k = 0; k < D_IN; k += 4) {
        const int kk = k + 2 * half;
        v2f a;
        a.x = xrow[kk];
        a.y = xrow[kk + 1];
        v2f b;
        b.x = wcol[(size_t)kk * D_OUT];
        b.y = wcol[(size_t)(kk + 1) * D_OUT];
        c = __builtin_amdgcn_wmma_f32_16x16x4_f32(
                /*neg_a=*/false, a, /*neg_b=*/false, b,
                /*c_mod=*/(short)0, c, /*reuse_a=*/false, /*reuse_b=*/false);
    }

    const float bn = bias[n0 + lm];
    #pragma unroll
    for (int j = 0; j < 8; ++j) {
        const int row = m0 + j + 8 * half;
        float v = c[j] + bn;
        h[(size_t)row * D_OUT + n0 + lm] = v > 0.0f ? v : 0.0f;
    }
}

// ------- Kernel 2: per-block partial sum / sumsq of e = 0.5*(h[r]+h[c]) -----
// 256 threads: tid&15 = column-group (4 cols, float4), tid>>4 = edge slot.
// Deterministic: LDS tree reduction, per-block partials (no float atomics).
__global__ __launch_bounds__(256)
void EdgeInit_edge_reduce(const float4* __restrict__ h4,
                          const int* __restrict__ ei,
                          int E,
                          float* __restrict__ partials) {
    __shared__ float lds_s[256 * 4];
    __shared__ float lds_q[256 * 4];
    const int tid = threadIdx.x;
    const int cg  = tid & 15;
    const int es  = tid >> 4;

    float sx = 0.f, sy = 0.f, sz = 0.f, sw = 0.f;
    float qx = 0.f, qy = 0.f, qz = 0.f, qw = 0.f;

    const int stride = gridDim.x * 16;
    for (int e = blockIdx.x * 16 + es; e < E; e += stride) {
        const int r = ei[e];
        const int c = ei[E + e];
        const float4 hr = h4[(size_t)r * 16 + cg];
        const float4 hc = h4[(size_t)c * 16 + cg];
        const float vx = 0.5f * (hr.x + hc.x);
        const float vy = 0.5f * (hr.y + hc.y);
        const float vz = 0.5f * (hr.z + hc.z);
        const float vw = 0.5f * (hr.w + hc.w);
        sx += vx; sy += vy; sz += vz; sw += vw;
        qx += vx * vx; qy += vy * vy; qz += vz * vz; qw += vw * vw;
    }

    lds_s[tid * 4 + 0] = sx; lds_s[tid * 4 + 1] = sy;
    lds_s[tid * 4 + 2] = sz; lds_s[tid * 4 + 3] = sw;
    lds_q[tid * 4 + 0] = qx; lds_q[tid * 4 + 1] = qy;
    lds_q[tid * 4 + 2] = qz; lds_q[tid * 4 + 3] = qw;
    __syncthreads();

    if (es == 0) {  // tid 0..15: one thread per column group
        float S0 = 0.f, S1 = 0.f, S2 = 0.f, S3 = 0.f;
        float Q0 = 0.f, Q1 = 0.f, Q2 = 0.f, Q3 = 0.f;
        #pragma unroll
        for (int j = 0; j < 16; ++j) {
            const int base = (j * 16 + cg) * 4;
            S0 += lds_s[base + 0]; S1 += lds_s[base + 1];
            S2 += lds_s[base + 2]; S3 += lds_s[base + 3];
            Q0 += lds_q[base + 0]; Q1 += lds_q[base + 1];
            Q2 += lds_q[base + 2]; Q3 += lds_q[base + 3];
        }
        float* p = partials + (size_t)blockIdx.x * 128;
        p[cg * 4 + 0] = S0; p[cg * 4 + 1] = S1;
        p[cg * 4 + 2] = S2; p[cg * 4 + 3] = S3;
        p[64 + cg * 4 + 0] = Q0; p[64 + cg * 4 + 1] = Q1;
        p[64 + cg * 4 + 2] = Q2; p[64 + cg * 4 + 3] = Q3;
    }
}

// ---- Kernel 3: reduce partials -> per-column affine coefficients a,b -------
__global__ __launch_bounds__(128)
void EdgeInit_finalize(const float* __restrict__ partials, int nb,
                       const float* __restrict__ gamma,
                       const float* __restrict__ beta,
                       float invE,
                       float* __restrict__ coefA,
                       float* __restrict__ coefB) {
    __shared__ float tot[128];
    const int tid = threadIdx.x;
    float s = 0.f;
    for (int b = 0; b < nb; ++b) s += partials[(size_t)b * 128 + tid];
    tot[tid] = s;
    __syncthreads();
    if (tid < 64) {
        const float mean = tot[tid] * invE;
        const float var  = tot[64 + tid] * invE - mean * mean;
        const float inv  = rsqrtf(var + 1e-5f);
        const float a    = inv * gamma[tid];
        coefA[tid] = a;
        coefB[tid] = beta[tid] - mean * a;
    }
}

// ---- Kernel 4: recompute e from L2-resident h, apply affine, write out -----
__global__ __launch_bounds__(256)
void EdgeInit_edge_norm(const float4* __restrict__ h4,
                        const int* __restrict__ ei,
                        int E,
                        const float4* __restrict__ cA4,
                        const float4* __restrict__ cB4,
                        float4* __restrict__ out4) {
    const int tid = threadIdx.x;
    const int cg  = tid & 15;
    const int es  = tid >> 4;
    const float4 a = cA4[cg];
    const float4 b = cB4[cg];

    const int stride = gridDim.x * 16;
    for (int e = blockIdx.x * 16 + es; e < E; e += stride) {
        if (e + stride < E) {
            __builtin_prefetch(&ei[e + stride], 0, 1);       // global_prefetch_b8
            __builtin_prefetch(&ei[E + e + stride], 0, 1);
        }
        const int r = ei[e];
        const int c = ei[E + e];
        const float4 hr = h4[(size_t)r * 16 + cg];
        const float4 hc = h4[(size_t)c * 16 + cg];
        float4 v;
        v.x = fmaf(0.5f * (hr.x + hc.x), a.x, b.x);
        v.y = fmaf(0.5f * (hr.y + hc.y), a.y, b.y);
        v.z = fmaf(0.5f * (hr.z + hc.z), a.z, b.z);
        v.w = fmaf(0.5f * (hr.w + hc.w), a.w, b.w);
        out4[(size_t)e * 16 + cg] = v;
    }
}

// ---------------------------------------------------------------------------
extern "C" void kernel_launch(void* const* d_in, const int* in_sizes, int n_in,
                              void* d_out, int out_size, void* d_ws, size_t ws_size,
                              hipStream_t stream) {
    const float* x     = (const float*)d_in[0];
    const int*   ei    = (const int*)  d_in[1];   // [2, E] indices
    const float* W     = (const float*)d_in[3];   // [256, 64] row-major
    const float* bias  = (const float*)d_in[4];
    const float* gamma = (const float*)d_in[5];
    const float* beta  = (const float*)d_in[6];

    const int N = in_sizes[0] / D_IN;             // 50000
    const int E = in_sizes[1] / 2;                // 800000

    // Workspace layout (floats): h | partials | coefA | coefB
    float* h        = (float*)d_ws;
    float* partials = h + (size_t)N * D_OUT;
    float* coefA    = partials + (size_t)NB_RED * 128;
    float* coefB    = coefA + 64;

    // 1) GEMM + bias + relu (N=50000 is an exact multiple of 16)
    dim3 ggrid(N / 16, D_OUT / 16);
    EdgeInit_gemm_relu<<<ggrid, 32, 0, stream>>>(x, W, bias, h);

    // 2) Deterministic per-block partial reduction of sum / sumsq
    EdgeInit_edge_reduce<<<NB_RED, 256, 0, stream>>>(
        (const float4*)h, ei, E, partials);

    // 3) Fold partials into per-column affine coefficients
    EdgeInit_finalize<<<1, 128, 0, stream>>>(
        partials, NB_RED, gamma, beta, 1.0f / (float)E, coefA, coefB);

    // 4) Recompute e from L2-resident h, apply affine, write output
    EdgeInit_edge_norm<<<4096, 256, 0, stream>>>(
        (const float4*)h, ei, E,
        (const float4*)coefA, (const float4*)coefB, (float4*)d_out);
}